// GCN_58136677319350
// MI455X (gfx1250) — compile-verified
//
#include <hip/hip_runtime.h>

// Probe results (gfx1250 toolchain): HAVE global_load_async_to_lds_b8/b64/b128,
// s_wait_asynccnt, s_wait_tensorcnt, tensor_load_to_lds, cluster_load_b32, etc.
// b64 async builtin signature: (v2i AS1*, v2i AS3*, imm offset, imm cpol).

#if __has_builtin(__builtin_amdgcn_global_load_async_to_lds_b64)
#define GCN_HAVE_ASYNC_B64 1
#else
#define GCN_HAVE_ASYNC_B64 0
#endif
#if __has_builtin(__builtin_amdgcn_s_wait_asynccnt)
#define GCN_WAIT_ASYNC(n) do { __builtin_amdgcn_s_wait_asynccnt(n); asm volatile("" ::: "memory"); } while (0)
#else
#define GCN_WAIT_ASYNC(n) do { asm volatile("s_wait_asynccnt %0" :: "i"(n) : "memory"); } while (0)
#endif

typedef int gcn_v2i __attribute__((ext_vector_type(2)));
typedef __attribute__((address_space(1))) gcn_v2i gcn_v2i_as1;
typedef __attribute__((address_space(3))) gcn_v2i gcn_v2i_as3;

#if GCN_HAVE_ASYNC_B64
__device__ __forceinline__ void async_row_b64(const float* gsrc, float* ldst) {
  // global (AS1) -> LDS (AS3), 8 bytes per lane => one 256B feature row per wave32
  __builtin_amdgcn_global_load_async_to_lds_b64(
      (gcn_v2i_as1*)(gcn_v2i*)(float*)gsrc,
      (gcn_v2i_as3*)(gcn_v2i*)ldst,
      /*offset=*/0, /*cpol=*/0);
}
#endif

#define P_DROP_F 0.3f
#define KEEP_SCALE (1.0f / 0.7f)

__device__ __forceinline__ float drop1(float x, float uu) {
  return (uu > P_DROP_F) ? x * KEEP_SCALE : 0.0f;
}

// ---------------------------------------------------------------------------
// Edge scatter (first in file so the disasm snippet shows the async pipeline).
// One wave32 per edge per iteration; lane l owns features [2l, 2l+1].
//   B[rows[e], :] += vals[e] * A[cols[e], :]
// The 256B gather row is staged global->LDS with async loads, 4-deep ring.
// ---------------------------------------------------------------------------
__global__ void __launch_bounds__(256)
GCN_scatter(const float* __restrict__ A, const float* __restrict__ vals,
            const int* __restrict__ rows, const int* __restrict__ cols,
            float* __restrict__ B, int E) {
  const int lane = threadIdx.x & 31;
  const int wiw  = threadIdx.x >> 5;                       // wave in workgroup (0..7)
  const int wave = blockIdx.x * (blockDim.x >> 5) + wiw;
  const int s    = gridDim.x * (blockDim.x >> 5);          // wave stride over edges

#if GCN_HAVE_ASYNC_B64
  __shared__ __align__(16) float lds[8][4][64];            // 4-deep ring per wave (8KB)
  // prologue: stage up to 3 rows ahead (branches are wave-uniform)
  #pragma unroll
  for (int b = 0; b < 3; ++b) {
    int ep = wave + b * s;
    if (ep < E) {
      int eu = __builtin_amdgcn_readfirstlane(ep);
      async_row_b64(A + cols[eu] * 64 + lane * 2, &lds[wiw][b][lane * 2]);
    }
  }
  int buf = 0;
  for (int e = wave; e < E; e += s, buf = (buf + 1) & 3) {
    // Issue lookahead row e+3s into the ring slot consumed 1 iteration ago,
    // then wait until the oldest outstanding load (row e) has landed in LDS.
    // Async loads complete in order; the wait immediate equals the number of
    // younger loads still allowed in flight.
    if (e + 3 * s < E) {
      int eu = __builtin_amdgcn_readfirstlane(e + 3 * s);
      async_row_b64(A + cols[eu] * 64 + lane * 2, &lds[wiw][(buf + 3) & 3][lane * 2]);
      GCN_WAIT_ASYNC(3);
    } else if (e + 2 * s < E) {
      GCN_WAIT_ASYNC(2);
    } else if (e + s < E) {
      GCN_WAIT_ASYNC(1);
    } else {
      GCN_WAIT_ASYNC(0);
    }
    const int eu = __builtin_amdgcn_readfirstlane(e);
    const float v = vals[eu];
    const int   r = rows[eu] * 64 + lane * 2;
    const float2 x = *(const float2*)&lds[wiw][buf][lane * 2];
    atomicAdd(&B[r],     v * x.x);
    atomicAdd(&B[r + 1], v * x.y);
  }
#else
  for (int e = wave; e < E; e += s) {
    const int eu = __builtin_amdgcn_readfirstlane(e);
    const float v = vals[eu];
    const float2 x = *(const float2*)(A + cols[eu] * 64 + lane * 2);
    const int r = rows[eu] * 64 + lane * 2;
    atomicAdd(&B[r],     v * x.x);
    atomicAdd(&B[r + 1], v * x.y);
  }
#endif
}

// Layer 0 fused prep:  A = h*keep(u0);  B = 0;  out = h   (unscaled running sum)
__global__ void __launch_bounds__(256)
GCN_prep_first(const float4* __restrict__ h, const float4* __restrict__ u,
               float4* __restrict__ A, float4* __restrict__ B,
               float4* __restrict__ out, int n4) {
  int i = blockIdx.x * blockDim.x + threadIdx.x;
  if (i >= n4) return;
  float4 x = h[i];
  float4 uu = u[i];
  float4 a;
  a.x = drop1(x.x, uu.x); a.y = drop1(x.y, uu.y);
  a.z = drop1(x.z, uu.z); a.w = drop1(x.w, uu.w);
  A[i] = a;
  B[i] = make_float4(0.f, 0.f, 0.f, 0.f);
  out[i] = x;
}

// Mid-layer fused prep: reads B (previous SpMM result), writes A = B*keep(u_l),
// accumulates out += B, then zeroes B for the next scatter (same index -> no race).
__global__ void __launch_bounds__(256)
GCN_prep_mid(const float4* __restrict__ u, float4* __restrict__ A,
             float4* __restrict__ B, float4* __restrict__ out, int n4) {
  int i = blockIdx.x * blockDim.x + threadIdx.x;
  if (i >= n4) return;
  float4 x = B[i];
  float4 uu = u[i];
  float4 a;
  a.x = drop1(x.x, uu.x); a.y = drop1(x.y, uu.y);
  a.z = drop1(x.z, uu.z); a.w = drop1(x.w, uu.w);
  A[i] = a;
  float4 o = out[i];
  o.x += x.x; o.y += x.y; o.z += x.z; o.w += x.w;
  out[i] = o;
  B[i] = make_float4(0.f, 0.f, 0.f, 0.f);
}

// Finalize: out = (out + B) * 0.25   (mean over the 4 representations)
__global__ void __launch_bounds__(256)
GCN_finalize(const float4* __restrict__ B, float4* __restrict__ out, int n4) {
  int i = blockIdx.x * blockDim.x + threadIdx.x;
  if (i >= n4) return;
  float4 x = B[i];
  float4 o = out[i];
  o.x = (o.x + x.x) * 0.25f; o.y = (o.y + x.y) * 0.25f;
  o.z = (o.z + x.z) * 0.25f; o.w = (o.w + x.w) * 0.25f;
  out[i] = o;
}

extern "C" void kernel_launch(void* const* d_in, const int* in_sizes, int n_in,
                              void* d_out, int out_size, void* d_ws, size_t ws_size,
                              hipStream_t stream) {
  const float* h    = (const float*)d_in[0];
  const float* vals = (const float*)d_in[1];
  const float* u    = (const float*)d_in[2];
  const int*   rows = (const int*)d_in[3];
  const int*   cols = (const int*)d_in[4];
  float*       out  = (float*)d_out;

  const int ND = in_sizes[0];     // N*D = 6,400,000
  const int E  = in_sizes[1];     // 1,600,000
  const int n4 = ND / 4;

  float* A = (float*)d_ws;        // dropped features  (25.6 MB)
  float* B = A + ND;              // SpMM accumulator  (25.6 MB)

  const int PB = (n4 + 255) / 256;
  const int SCAT_BLOCKS = 4096;   // 32768 waves, grid-strided over edges

  // layer 0
  GCN_prep_first<<<PB, 256, 0, stream>>>((const float4*)h, (const float4*)u,
                                         (float4*)A, (float4*)B, (float4*)out, n4);
  GCN_scatter<<<SCAT_BLOCKS, 256, 0, stream>>>(A, vals, rows, cols, B, E);
  // layer 1
  GCN_prep_mid<<<PB, 256, 0, stream>>>((const float4*)(u + (size_t)ND),
                                       (float4*)A, (float4*)B, (float4*)out, n4);
  GCN_scatter<<<SCAT_BLOCKS, 256, 0, stream>>>(A, vals, rows, cols, B, E);
  // layer 2
  GCN_prep_mid<<<PB, 256, 0, stream>>>((const float4*)(u + (size_t)2 * ND),
                                       (float4*)A, (float4*)B, (float4*)out, n4);
  GCN_scatter<<<SCAT_BLOCKS, 256, 0, stream>>>(A, vals, rows, cols, B, E);
  // mean over {h, cur1, cur2, cur3}
  GCN_finalize<<<PB, 256, 0, stream>>>((const float4*)B, (float4*)out, n4);
}